// FastCrossMessageTokenAttention_5549097746991
// MI455X (gfx1250) — compile-verified
//
#include <hip/hip_runtime.h>
#include <hip/hip_bf16.h>
#include <math.h>

typedef __attribute__((ext_vector_type(2))) float v2f;
typedef __attribute__((ext_vector_type(8))) float v8f;

#define T_TOK   16384
#define MSGS    256
#define LTOK    64
#define HID     256
#define NHEAD   4
#define HD      64
#define KKSEL   6
#define SSEL    (MSGS * KKSEL)     // 1536
#define KT      10
#define NEGF    (-1.0e30f)

// ---------------------------------------------------------------------------
// fp32 WMMA GEMM: Y[M x N] = X[M x 256] @ W[N x 256]^T + bias, opt relu.
// Each wave computes a 16x32 strip: one A fragment feeds TWO independent
// v_wmma_f32_16x16x4_f32 accumulators per k-step (2x less A traffic than a
// single tile), while the low register footprint (2 accumulators + 3 load
// streams, unroll 8) lets the scheduler keep ~24 loads in flight with
// staggered s_wait_loadcnt instead of serializing load->wait->wmma.
// A (16x4): lanes 0-15 rows M=0..15 hold K={0,1}; lanes 16-31 hold K={2,3}.
// B (4x16): lanes 0-15 cols N=0..15 hold K={0,1}; lanes 16-31 hold K={2,3}.
// C (16x16): VGPR i, lanes 0-15 -> M=i, lanes 16-31 -> M=i+8; N = lane&15.
// Requires Ncols % 32 == 0.
// ---------------------------------------------------------------------------
__global__ __launch_bounds__(256)
void gemm_wmma_f32(const float* __restrict__ X, const float* __restrict__ W,
                   const float* __restrict__ bias, float* __restrict__ Y,
                   int Mrows, int Ncols, int relu_flag) {
  const int KDIM = 256;
  int wave = blockIdx.x * (blockDim.x >> 5) + (threadIdx.x >> 5);
  int lane = threadIdx.x & 31;
  int stripsN = Ncols >> 5;                   // 32-wide strips
  int totalStrips = (Mrows >> 4) * stripsN;
  if (wave >= totalStrips) return;
  int tm = (wave / stripsN) << 4;
  int tn = (wave % stripsN) << 5;
  int r  = lane & 15;
  int kh = (lane >> 4) << 1;                  // 0 or 2
  const float* xrow = X + (size_t)(tm + r) * KDIM + kh;
  const float* w0 = W + (size_t)(tn      + r) * KDIM + kh;
  const float* w1 = W + (size_t)(tn + 16 + r) * KDIM + kh;
  v8f c0 = {}, c1 = {};
#pragma unroll 8
  for (int k = 0; k < KDIM; k += 4) {
    v2f a  = *(const v2f*)(xrow + k);
    v2f b0 = *(const v2f*)(w0 + k);
    v2f b1 = *(const v2f*)(w1 + k);
    c0 = __builtin_amdgcn_wmma_f32_16x16x4_f32(false, a, false, b0, (short)0, c0, false, false);
    c1 = __builtin_amdgcn_wmma_f32_16x16x4_f32(false, a, false, b1, (short)0, c1, false, false);
  }
  int rowbase = tm + ((lane >> 4) << 3);
  int cbase = tn + (lane & 15);
#pragma unroll
  for (int t = 0; t < 2; ++t) {
    const v8f& c = (t == 0) ? c0 : c1;
    int col = cbase + t * 16;
    float bv = bias ? bias[col] : 0.0f;
#pragma unroll
    for (int i = 0; i < 8; ++i) {
      float v = c[i] + bv;
      if (relu_flag) v = fmaxf(v, 0.0f);
      Y[(size_t)(rowbase + i) * Ncols + col] = v;
    }
  }
}

// imp[t] = bi2 + dot(H[t,:128], Wi2[:128])
__global__ void imp_kernel(const float* __restrict__ H, const float* __restrict__ Wi2,
                           const float* __restrict__ bi2, float* __restrict__ imp) {
  int t = blockIdx.x * blockDim.x + threadIdx.x;
  if (t >= T_TOK) return;
  float s = bi2[0];
  const float* hr = H + (size_t)t * 128;
#pragma unroll 8
  for (int j = 0; j < 128; ++j) s += hr[j] * Wi2[j];
  imp[t] = s;
}

// Top-6 per message (descending, first-occurrence ties like jax top_k)
__global__ void topk6_kernel(const float* __restrict__ imp, int* __restrict__ selected) {
  int m = blockIdx.x * blockDim.x + threadIdx.x;
  if (m >= MSGS) return;
  const float* v = imp + m * LTOK;
  int chosen[KKSEL];
#pragma unroll
  for (int c = 0; c < KKSEL; ++c) {
    float best = -3.4e38f; int bi = 0;
    for (int j = 0; j < LTOK; ++j) {
      bool used = false;
#pragma unroll
      for (int p = 0; p < KKSEL; ++p) used |= (p < c) && (chosen[p] == j);
      float val = used ? -3.4e38f : v[j];
      if (val > best) { best = val; bi = j; }
    }
    chosen[c] = bi;
    selected[m * KKSEL + c] = m * LTOK + bi;
  }
}

__global__ void gather_rows_kernel(const float* __restrict__ X,
                                   const int* __restrict__ selected,
                                   float* __restrict__ Y) {
  int i = blockIdx.x * blockDim.x + threadIdx.x;
  if (i >= SSEL * HID) return;
  int s = i >> 8, d = i & 255;
  Y[i] = X[(size_t)selected[s] * HID + d];
}

// ---------------------------------------------------------------------------
// Scores via WMMA + streaming top-10 + softmax + V gather.
// grid = (SSEL/16, NHEAD), block = 256 (8 waves).
// ---------------------------------------------------------------------------
__global__ __launch_bounds__(256)
void attn_topk_kernel(const float* __restrict__ Q, const float* __restrict__ Kmat,
                      const float* __restrict__ Vmat, const int* __restrict__ selected,
                      const int* __restrict__ batch_indices,
                      float* __restrict__ top_scores_ws, float* __restrict__ attended) {
  int q0 = blockIdx.x << 4;
  int h  = blockIdx.y;
  int tid = threadIdx.x;
  int lane = tid & 31, wave = tid >> 5;

  __shared__ float lds_sc[8][16][16];
  __shared__ float lds_ts[8][16][KT];
  __shared__ int   lds_ti[8][16][KT];
  __shared__ float lds_w[16][KT];
  __shared__ int   lds_idx[16][KT];
  __shared__ int   lds_batch[MSGS];

  for (int i = tid; i < MSGS; i += blockDim.x) lds_batch[i] = batch_indices[i];
  __syncthreads();

  // Preload A fragments for all 16 k-steps (Q tile, head h): 32 floats/lane
  int r  = lane & 15;
  int kh = (lane >> 4) << 1;
  v2f a[16];
  const float* qrow = Q + (size_t)(q0 + r) * HID + h * HD + kh;
#pragma unroll
  for (int kk = 0; kk < 16; ++kk) a[kk] = *(const v2f*)(qrow + kk * 4);

  int myrow = lane & 15;
  int q_msg = selected[q0 + myrow] >> 6;     // token/64
  int q_bat = lds_batch[q_msg];

  float ts[KT]; int ti[KT];
#pragma unroll
  for (int k = 0; k < KT; ++k) { ts[k] = NEGF; ti[k] = k; }

  const float scale = 0.125f;                // 1/sqrt(HD)

  for (int tile = wave; tile < (T_TOK >> 4); tile += 8) {
    int tb = tile << 4;
    int tmsg = tb >> 6;                      // whole tile lies in one message
    const float* krow = Kmat + (size_t)(tb + r) * HID + h * HD + kh;
    v8f c = {};
#pragma unroll
    for (int kk = 0; kk < 16; ++kk) {
      v2f b = *(const v2f*)(krow + kk * 4);
      c = __builtin_amdgcn_wmma_f32_16x16x4_f32(false, a[kk], false, b, (short)0,
                                                c, false, false);
    }
    int rowb = (lane >> 4) << 3;
#pragma unroll
    for (int i = 0; i < 8; ++i) lds_sc[wave][rowb + i][lane & 15] = c[i] * scale;
    __asm__ volatile("s_wait_dscnt 0x0" ::: "memory");   // wave-local LDS visibility

    if (lane < 16) {
      bool valid = (tmsg != q_msg) && (lds_batch[tmsg] == q_bat);
      if (valid) {
#pragma unroll
        for (int j = 0; j < 16; ++j) {
          float s = lds_sc[wave][myrow][j];
          if (s > ts[KT - 1]) {              // sorted insert, fully unrolled
            ts[KT - 1] = s; ti[KT - 1] = tb + j;
#pragma unroll
            for (int p = KT - 1; p > 0; --p) {
              if (ts[p] > ts[p - 1]) {
                float tf = ts[p]; ts[p] = ts[p - 1]; ts[p - 1] = tf;
                int   tx = ti[p]; ti[p] = ti[p - 1]; ti[p - 1] = tx;
              }
            }
          }
        }
      }
    }
  }

  if (lane < 16) {
#pragma unroll
    for (int k = 0; k < KT; ++k) { lds_ts[wave][myrow][k] = ts[k]; lds_ti[wave][myrow][k] = ti[k]; }
  }
  __syncthreads();

  if (tid < 16) {
    int m = tid;
    float fs[KT]; int fi[KT];
#pragma unroll
    for (int k = 0; k < KT; ++k) { fs[k] = NEGF; fi[k] = k; }
    for (int w = 0; w < 8; ++w) {
#pragma unroll
      for (int k = 0; k < KT; ++k) {
        float s = lds_ts[w][m][k];
        int ix  = lds_ti[w][m][k];
        if (s > fs[KT - 1]) {
          fs[KT - 1] = s; fi[KT - 1] = ix;
#pragma unroll
          for (int p = KT - 1; p > 0; --p) {
            if (fs[p] > fs[p - 1]) {
              float tf = fs[p]; fs[p] = fs[p - 1]; fs[p - 1] = tf;
              int   tx = fi[p]; fi[p] = fi[p - 1]; fi[p - 1] = tx;
            }
          }
        }
      }
    }
    float mx = fs[0];
    float ex[KT]; float sum = 0.0f;
#pragma unroll
    for (int k = 0; k < KT; ++k) { ex[k] = expf(fs[k] - mx); sum += ex[k]; }
    float inv = 1.0f / sum;
    int b = q0 + m;
#pragma unroll
    for (int k = 0; k < KT; ++k) {
      top_scores_ws[((size_t)b * NHEAD + h) * KT + k] = fs[k];
      lds_w[m][k] = ex[k] * inv;
      lds_idx[m][k] = fi[k];
    }
  }
  __syncthreads();

  // attended[b, h*64 + d] = sum_k w[k] * V[idx[k], h, d] ; 16 rows x 64 dims
  {
    int m = tid >> 4;
    int d0 = (tid & 15) << 2;
    float a0 = 0, a1 = 0, a2 = 0, a3 = 0;
#pragma unroll
    for (int k = 0; k < KT; ++k) {
      float wt = lds_w[m][k];
      const float* vp = Vmat + (size_t)lds_idx[m][k] * HID + h * HD + d0;
      a0 += wt * vp[0]; a1 += wt * vp[1]; a2 += wt * vp[2]; a3 += wt * vp[3];
    }
    float* op = attended + (size_t)(q0 + m) * HID + h * HD + d0;
    op[0] = a0; op[1] = a1; op[2] = a2; op[3] = a3;
  }
}

__global__ void copy_out_kernel(const float* __restrict__ X, float* __restrict__ out) {
  int i = blockIdx.x * blockDim.x + threadIdx.x;
  if (i < T_TOK * HID) out[i] = X[i];
}

__global__ void scatter_add_kernel(const float* __restrict__ updates,
                                   const int* __restrict__ selected,
                                   float* __restrict__ out) {
  int i = blockIdx.x * blockDim.x + threadIdx.x;
  if (i >= SSEL * HID) return;
  int s = i >> 8, d = i & 255;
  out[(size_t)selected[s] * HID + d] += updates[i];   // selected rows unique
}

__global__ void avg_scores_kernel(const float* __restrict__ tsw, float* __restrict__ oa) {
  int i = blockIdx.x * blockDim.x + threadIdx.x;
  if (i >= SSEL * KT) return;
  int b = i / KT, k = i % KT;
  float s = 0.0f;
#pragma unroll
  for (int h = 0; h < NHEAD; ++h) s += tsw[((size_t)b * NHEAD + h) * KT + k];
  oa[i] = s * 0.25f;
}

extern "C" void kernel_launch(void* const* d_in, const int* in_sizes, int n_in,
                              void* d_out, int out_size, void* d_ws, size_t ws_size,
                              hipStream_t stream) {
  const float* X   = (const float*)d_in[0];
  const int*   bat = (const int*)d_in[2];
  const float* Wq  = (const float*)d_in[3];
  const float* bq  = (const float*)d_in[4];
  const float* Wk  = (const float*)d_in[5];
  const float* bk  = (const float*)d_in[6];
  const float* Wv  = (const float*)d_in[7];
  const float* bv  = (const float*)d_in[8];
  const float* Wi1 = (const float*)d_in[9];
  const float* bi1 = (const float*)d_in[10];
  const float* Wi2 = (const float*)d_in[11];
  const float* bi2 = (const float*)d_in[12];
  const float* Wo  = (const float*)d_in[13];
  const float* bo  = (const float*)d_in[14];

  float* ws = (float*)d_ws;
  float* Kmat = ws;                               // 16384*256
  float* Vmat = Kmat + (size_t)T_TOK * HID;       // 16384*256
  float* H    = Vmat + (size_t)T_TOK * HID;       // 16384*128
  float* imp  = H + (size_t)T_TOK * 128;          // 16384
  float* Qin  = imp + T_TOK;                      // 1536*256
  float* Qp   = Qin + (size_t)SSEL * HID;         // 1536*256
  float* att  = Qp + (size_t)SSEL * HID;          // 1536*256
  float* upd  = att + (size_t)SSEL * HID;         // 1536*256
  float* tsw  = upd + (size_t)SSEL * HID;         // 1536*4*10
  int*   sel  = (int*)(tsw + (size_t)SSEL * NHEAD * KT);  // 1536 ints

  float* out_updated = (float*)d_out;             // 16384*256
  float* out_avg     = out_updated + (size_t)T_TOK * HID; // 1536*10

  // K / V / hidden projections (WMMA GEMMs, 16x32 strip per wave, 8 waves/blk)
  {
    int strips = (T_TOK / 16) * (HID / 32);       // 8192
    gemm_wmma_f32<<<(strips + 7) / 8, 256, 0, stream>>>(X, Wk, bk, Kmat, T_TOK, HID, 0);
    gemm_wmma_f32<<<(strips + 7) / 8, 256, 0, stream>>>(X, Wv, bv, Vmat, T_TOK, HID, 0);
    int stripsH = (T_TOK / 16) * (128 / 32);      // 4096
    gemm_wmma_f32<<<(stripsH + 7) / 8, 256, 0, stream>>>(X, Wi1, bi1, H, T_TOK, 128, 1);
  }
  imp_kernel<<<(T_TOK + 255) / 256, 256, 0, stream>>>(H, Wi2, bi2, imp);
  topk6_kernel<<<1, 256, 0, stream>>>(imp, sel);
  gather_rows_kernel<<<(SSEL * HID + 255) / 256, 256, 0, stream>>>(X, sel, Qin);
  {
    int strips = (SSEL / 16) * (HID / 32);        // 768
    gemm_wmma_f32<<<(strips + 7) / 8, 256, 0, stream>>>(Qin, Wq, bq, Qp, SSEL, HID, 0);
  }
  attn_topk_kernel<<<dim3(SSEL / 16, NHEAD), 256, 0, stream>>>(Qp, Kmat, Vmat, sel,
                                                               bat, tsw, att);
  {
    int strips = (SSEL / 16) * (HID / 32);
    gemm_wmma_f32<<<(strips + 7) / 8, 256, 0, stream>>>(att, Wo, bo, upd, SSEL, HID, 0);
  }
  copy_out_kernel<<<(T_TOK * HID + 255) / 256, 256, 0, stream>>>(X, out_updated);
  scatter_add_kernel<<<(SSEL * HID + 255) / 256, 256, 0, stream>>>(upd, sel, out_updated);
  avg_scores_kernel<<<(SSEL * KT + 255) / 256, 256, 0, stream>>>(tsw, out_avg);
}